// GNNParser_63960652972177
// MI455X (gfx1250) — compile-verified
//
#include <hip/hip_runtime.h>
#include <hip/hip_bf16.h>

#define BB   64
#define SS   512
#define SM1  511
#define EMBD 1024
#define ADP  512      // padded ARC_D (500 -> 512)
#define TDP  128      // padded TAG_D (100 -> 128)
#define NARC 500
#define NTAG 100
#define BS   (BB * SS)

typedef __attribute__((ext_vector_type(2))) float v2f;
typedef __attribute__((ext_vector_type(8))) float v8f;

// ---------------------------------------------------------------------------
// Specialized f32 WMMA GEMM:  C = act( (A [+A2]) * B  [+ bias] [+ C] )
// One 16x16 C tile per wave via V_WMMA_F32_16X16X4_F32; 256 thr = 8 waves
// = 32x64 block tile. All structural options are template params so every
// variant compiles to a branch-free inner loop (bounds via clamp + select).
//
// AOPT: 0 = plain (no K bound), 1 = K bound + virtual ones col at k==KA,
//       2 = K bound only, 3 = sentinel mode ([head_sentinel; input] rows).
// UNR : K-step = 4*UNR (K4 must be a multiple of it).
// BIASM: 0 none, 1 vector bias[n] (bounded by biasN), 2 scalar bias[0].
// ACT : 0 none, 1 ELU, 2 tanh.   BETA: accumulate into existing C.
// ---------------------------------------------------------------------------
template<int UNR, int AOPT, bool TA, bool A2F, bool TB, bool KBCHK, bool BONES,
         bool NBCHK, int BIASM, int ACT, bool BETA>
__global__ __launch_bounds__(256)
void wmma_gemm(const float* __restrict__ A, const float* __restrict__ A2,
               long lda, long sA, int KA,
               const float* __restrict__ Bm, long ldb, long sB, int KB, int NB,
               float* __restrict__ C, long ldc, long sC,
               const float* __restrict__ bias, int biasN, int K4,
               const float* __restrict__ sent, const float* __restrict__ xin)
{
    const int lane = threadIdx.x & 31;
    const int w    = threadIdx.x >> 5;
    const int half = lane >> 4;              // 0: lanes 0-15, 1: lanes 16-31
    const int r    = lane & 15;
    const int m0   = blockIdx.y * 32 + (w >> 2) * 16;
    const int n0   = blockIdx.x * 64 + (w & 3) * 16;
    const int bz   = blockIdx.z;

    const int am = m0 + r;                   // A row for this lane
    const int bn = n0 + r;                   // B/C column for this lane
    const bool nOK = !NBCHK || (bn < NB);
    const int  bnc = NBCHK ? (nOK ? bn : 0) : bn;

    const float* Ap;
    const float* A2p = nullptr;
    if (AOPT == 3) {
        const int bbi = am >> 9;             // / SS
        const int ss  = am & (SS - 1);       // % SS
        Ap = (ss == 0) ? sent : (xin + ((long)(bbi * SM1 + ss - 1)) * EMBD);
    } else {
        const float* Ab = A + (long)bz * sA;
        Ap = TA ? (Ab + am) : (Ab + (long)am * lda);
        if (A2F) A2p = A2 + (long)bz * sA + (TA ? (long)am : (long)am * lda);
    }
    const float* Bb = Bm + (long)bz * sB;
    const float* Bp = TB ? (Bb + (long)bnc * ldb) : (Bb + bnc);
    float*       Cb = C + (long)bz * sC;

    v8f acc = {0.f, 0.f, 0.f, 0.f, 0.f, 0.f, 0.f, 0.f};

    for (int k = 0; k < K4; k += 4 * UNR) {
        #pragma unroll
        for (int u = 0; u < UNR; ++u) {
            const int kk0 = k + 4 * u + 2 * half;
            const int kk1 = kk0 + 1;
            float a0, a1, b0, b1;

            // ---- A fragment ----
            if (AOPT == 0 || AOPT == 3) {
                if (!TA) {
                    a0 = Ap[kk0]; a1 = Ap[kk1];
                    if (A2F) { a0 += A2p[kk0]; a1 += A2p[kk1]; }
                } else {
                    a0 = Ap[(long)kk0 * lda]; a1 = Ap[(long)kk1 * lda];
                }
            } else {
                const int kc0 = kk0 < KA ? kk0 : 0;
                const int kc1 = kk1 < KA ? kk1 : 0;
                float r0, r1;
                if (!TA) { r0 = Ap[kc0]; r1 = Ap[kc1]; }
                else     { r0 = Ap[(long)kc0 * lda]; r1 = Ap[(long)kc1 * lda]; }
                a0 = (kk0 < KA) ? r0 : ((AOPT == 1 && kk0 == KA) ? 1.f : 0.f);
                a1 = (kk1 < KA) ? r1 : ((AOPT == 1 && kk1 == KA) ? 1.f : 0.f);
            }

            // ---- B fragment ----
            {
                const int kc0 = KBCHK ? (kk0 < KB ? kk0 : 0) : kk0;
                const int kc1 = KBCHK ? (kk1 < KB ? kk1 : 0) : kk1;
                float r0, r1;
                if (!TB) { r0 = Bp[(long)kc0 * ldb]; r1 = Bp[(long)kc1 * ldb]; }
                else     { r0 = Bp[kc0]; r1 = Bp[kc1]; }
                if (KBCHK) {
                    b0 = (kk0 < KB) ? r0 : ((BONES && kk0 == KB) ? 1.f : 0.f);
                    b1 = (kk1 < KB) ? r1 : ((BONES && kk1 == KB) ? 1.f : 0.f);
                } else { b0 = r0; b1 = r1; }
                if (NBCHK && !nOK) { b0 = 0.f; b1 = 0.f; }
            }

            v2f av; av.x = a0; av.y = a1;
            v2f bv; bv.x = b0; bv.y = b1;
            acc = __builtin_amdgcn_wmma_f32_16x16x4_f32(false, av, false, bv,
                                                        (short)0, acc, false, false);
        }
    }

    float badd = 0.f;
    if (BIASM == 1) {
        const int nc = bn < biasN ? bn : 0;
        const float bv2 = bias[nc];
        badd = (bn < biasN) ? bv2 : 0.f;
    } else if (BIASM == 2) {
        badd = bias[0];
    }

    #pragma unroll
    for (int i = 0; i < 8; ++i) {
        const long ci = (long)(m0 + i + 8 * half) * ldc + bn;
        float v = acc[i] + badd;
        if (BETA) v += Cb[ci];
        if (ACT == 1)      v = v > 0.f ? v : expm1f(v);   // ELU
        else if (ACT == 2) v = tanhf(v);
        Cb[ci] = v;
    }
}

// ---------------------------------------------------------------------------
// Row softmax (in place) + masked log-softmax + gathered NLL term.
// ---------------------------------------------------------------------------
__global__ __launch_bounds__(256)
void softmax_loss_kernel(float* __restrict__ arc, const int* __restrict__ mask,
                         const int* __restrict__ hidx, float* __restrict__ loss_acc)
{
    __shared__ float red[256];
    const int row = blockIdx.x;
    const int b = row >> 9;
    const int s = row & (SS - 1);
    float* p = arc + (long)row * SS;

    const int t0 = threadIdx.x;
    const int t1 = threadIdx.x + 256;
    float x0 = p[t0], x1 = p[t1];

    float fm0 = (t0 == 0) ? 1.f : (float)mask[b * SM1 + t0 - 1];
    float fm1 = (float)mask[b * SM1 + t1 - 1];
    float y0 = x0 + logf(fm0 + 1e-45f);
    float y1 = x1 + logf(fm1 + 1e-45f);

    red[threadIdx.x] = fmaxf(x0, x1); __syncthreads();
    for (int o = 128; o > 0; o >>= 1) {
        if (threadIdx.x < o) red[threadIdx.x] = fmaxf(red[threadIdx.x], red[threadIdx.x + o]);
        __syncthreads();
    }
    float rmax = red[0]; __syncthreads();
    float e0 = __expf(x0 - rmax), e1 = __expf(x1 - rmax);
    red[threadIdx.x] = e0 + e1; __syncthreads();
    for (int o = 128; o > 0; o >>= 1) {
        if (threadIdx.x < o) red[threadIdx.x] += red[threadIdx.x + o];
        __syncthreads();
    }
    float inv = 1.f / red[0]; __syncthreads();
    p[t0] = e0 * inv;
    p[t1] = e1 * inv;

    red[threadIdx.x] = fmaxf(y0, y1); __syncthreads();
    for (int o = 128; o > 0; o >>= 1) {
        if (threadIdx.x < o) red[threadIdx.x] = fmaxf(red[threadIdx.x], red[threadIdx.x + o]);
        __syncthreads();
    }
    float m2 = red[0]; __syncthreads();
    red[threadIdx.x] = __expf(y0 - m2) + __expf(y1 - m2); __syncthreads();
    for (int o = 128; o > 0; o >>= 1) {
        if (threadIdx.x < o) red[threadIdx.x] += red[threadIdx.x + o];
        __syncthreads();
    }
    float lse = m2 + logf(red[0]);

    if (s > 0) {
        int h = hidx[b * SM1 + s - 1];
        if (t0 == h)      atomicAdd(loss_acc, (y0 - lse) * fm0);
        else if (t1 == h) atomicAdd(loss_acc, (y1 - lse) * fm1);
    }
}

__global__ void mask_sum_kernel(const int* __restrict__ mask, int n, float* __restrict__ out)
{
    __shared__ float red[256];
    int i = blockIdx.x * 256 + threadIdx.x;
    red[threadIdx.x] = (i < n) ? (float)mask[i] : 0.f;
    __syncthreads();
    for (int o = 128; o > 0; o >>= 1) {
        if (threadIdx.x < o) red[threadIdx.x] += red[threadIdx.x + o];
        __syncthreads();
    }
    if (threadIdx.x == 0) atomicAdd(out, red[0]);
}

__global__ void finalize_kernel(const float* __restrict__ sc, float* __restrict__ out)
{
    int k = threadIdx.x;
    if (k < 3) out[k] = -sc[k] / sc[3];
}

__global__ void copy_tag_kernel(const float* __restrict__ src, float* __restrict__ dst, long n)
{
    long i = (long)blockIdx.x * blockDim.x + threadIdx.x;
    if (i < n) {
        long rrow = i / NTAG;
        int  c = (int)(i - rrow * NTAG);
        dst[i] = src[rrow * TDP + c];
    }
}

// ---------------------------------------------------------------------------
// Host-side wrappers (one per template instantiation)
// ---------------------------------------------------------------------------
static inline dim3 grd(int M, int Npad, int batch) { return dim3(Npad / 64, M / 32, batch); }

// proj: C = elu([sent;input] @ W + b), K = 1024 (exact), N bounded
static inline void g_proj(hipStream_t st, const float* W, int Nreal, const float* bvec,
                          float* Cb, int Npad, const float* sent, const float* xin)
{
    wmma_gemm<4, 3, false, false, false, false, false, true, 1, 1, false>
        <<<grd(BS, Npad, 1), 256, 0, st>>>(nullptr, nullptr, 0, 0, EMBD,
                                           W, Nreal, 0, EMBD, Nreal,
                                           Cb, Npad, 0, bvec, Nreal, EMBD, sent, xin);
}
// bilinear stage 1: T = [HA,1] @ Wk  (K4=504)
static inline void g_bil1(hipStream_t st, const float* HA, const float* Wk, float* T)
{
    wmma_gemm<2, 1, false, false, false, true, false, true, 0, 0, false>
        <<<grd(BS, ADP, 1), 256, 0, st>>>(HA, nullptr, ADP, 0, NARC,
                                          Wk, 501, 0, 501, 501,
                                          T, ADP, 0, nullptr, 0, 504, nullptr, nullptr);
}
// bilinear stage 2: P = T @ [DA,1]^T + b  (batched, K4=504)
static inline void g_bil2(hipStream_t st, const float* T, const float* DA,
                          float* P, const float* bscal)
{
    wmma_gemm<2, 2, false, false, true, true, true, false, 2, 0, false>
        <<<grd(SS, SS, BB), 256, 0, st>>>(T, nullptr, ADP, (long)SS * ADP, 501,
                                          DA, ADP, (long)SS * ADP, NARC, SS,
                                          P, SS, (long)SS * SS, bscal, 0, 504,
                                          nullptr, nullptr);
}
// batched O = P @ X (no checks, K = 512)
static inline void g_pX(hipStream_t st, const float* P, const float* X, float* O, int ld)
{
    wmma_gemm<4, 0, false, false, false, false, false, false, 0, 0, false>
        <<<grd(SS, ld, BB), 256, 0, st>>>(P, nullptr, SS, (long)SS * SS, 0,
                                          X, ld, (long)SS * ld, 0, 0,
                                          O, ld, (long)SS * ld, nullptr, 0, SS,
                                          nullptr, nullptr);
}
// batched O = P^T @ X (no checks, K = 512)
static inline void g_pXT(hipStream_t st, const float* P, const float* X, float* O, int ld)
{
    wmma_gemm<4, 0, true, false, false, false, false, false, 0, 0, false>
        <<<grd(SS, ld, BB), 256, 0, st>>>(P, nullptr, SS, (long)SS * SS, 0,
                                          X, ld, (long)SS * ld, 0, 0,
                                          O, ld, (long)SS * ld, nullptr, 0, SS,
                                          nullptr, nullptr);
}
// update pass 1: O = (A1 + A2) @ W
static inline void g_upd1(hipStream_t st, const float* A1p, const float* A2p,
                          const float* W, float* O, int ld, int Nreal)
{
    wmma_gemm<4, 0, false, true, false, true, false, true, 0, 0, false>
        <<<grd(BS, ld, 1), 256, 0, st>>>(A1p, A2p, ld, 0, 0,
                                         W, Nreal, 0, Nreal, Nreal,
                                         O, ld, 0, nullptr, 0, ld, nullptr, nullptr);
}
// update pass 2: O = tanh(O + X @ W)
static inline void g_upd2(hipStream_t st, const float* X, const float* W,
                          float* O, int ld, int Nreal)
{
    wmma_gemm<4, 0, false, false, false, true, false, true, 0, 2, true>
        <<<grd(BS, ld, 1), 256, 0, st>>>(X, nullptr, ld, 0, 0,
                                         W, Nreal, 0, Nreal, Nreal,
                                         O, ld, 0, nullptr, 0, ld, nullptr, nullptr);
}

template <typename T> static inline void swp(T*& a, T*& b) { T* t = a; a = b; b = t; }

extern "C" void kernel_launch(void* const* d_in, const int* in_sizes, int n_in,
                              void* d_out, int out_size, void* d_ws, size_t ws_size,
                              hipStream_t stream)
{
    (void)in_sizes; (void)n_in; (void)out_size; (void)ws_size;

    const float* input = (const float*)d_in[0];
    const int*   mask  = (const int*)d_in[2];
    const int*   hidx  = (const int*)d_in[3];
    const float* sent  = (const float*)d_in[5];
    const float* W_ha = (const float*)d_in[6],  *b_ha = (const float*)d_in[7];
    const float* W_da = (const float*)d_in[8],  *b_da = (const float*)d_in[9];
    const float* W_ht = (const float*)d_in[10], *b_ht = (const float*)d_in[11];
    const float* W_dt = (const float*)d_in[12], *b_dt = (const float*)d_in[13];
    const float* bilW = (const float*)d_in[14], *bilb = (const float*)d_in[15];
    const float* gHW  = (const float*)d_in[16], *gHB  = (const float*)d_in[17];
    const float* gDW  = (const float*)d_in[18], *gDB  = (const float*)d_in[19];
    const float* gHRW = (const float*)d_in[20], *gHRB = (const float*)d_in[21];
    const float* gDRW = (const float*)d_in[22], *gDRB = (const float*)d_in[23];

    const long SZA = (long)BS * ADP;
    const long SZT = (long)BS * TDP;
    const long SZP = (long)BB * SS * SS;

    float* ws  = (float*)d_ws;
    float* A0  = ws;
    float* A1  = A0 + SZA;
    float* D0  = A1 + SZA;
    float* D1  = D0 + SZA;
    float* P   = D1 + SZA;                 // arc scores / arc_p
    float* T   = P + SZP;                  // bilinear intermediate / dx / dr
    float* AC  = T + SZA;                  // hx / hr accumulators
    float* Th0 = AC + SZA;
    float* Th1 = Th0 + SZT;
    float* Td0 = Th1 + SZT;
    float* Td1 = Td0 + SZT;
    float* SC  = Td1 + SZT;                // [loss0,loss1,loss2,valid]

    hipMemsetAsync(SC, 0, 4 * sizeof(float), stream);
    mask_sum_kernel<<<(BB * SM1 + 255) / 256, 256, 0, stream>>>(mask, BB * SM1, SC + 3);

    // --- projections: feat = elu(x @ W + b), x = [sentinel; input] ---
    g_proj(stream, W_ha, NARC, b_ha, A0, ADP, sent, input);
    g_proj(stream, W_da, NARC, b_da, D0, ADP, sent, input);
    g_proj(stream, W_ht, NTAG, b_ht, Th0, TDP, sent, input);
    g_proj(stream, W_dt, NTAG, b_dt, Td0, TDP, sent, input);

    float *HA = A0, *HAo = A1, *DA = D0, *DAo = D1;
    float *HT = Th0, *HTo = Th1, *DT = Td0, *DTo = Td1;

    for (int k = 0; k < 3; ++k) {
        const float* Wk = bilW + (long)k * 501 * 501;

        g_bil1(stream, HA, Wk, T);                         // T = h1 @ Wk
        g_bil2(stream, T, DA, P, bilb + k);                // arc_s = T @ d1^T + b
        softmax_loss_kernel<<<BS, 256, 0, stream>>>(P, mask, hidx, SC + k);

        g_pXT(stream, P, DA, T, ADP);                      // dx = P^T @ DA
        g_pX (stream, P, HA, AC, ADP);                     // hx = P @ HA
        g_upd1(stream, AC, T, gHW, HAo, ADP, NARC);        // (hx+dx)@gHW
        g_upd2(stream, HA, gHB, HAo, ADP, NARC);           // + HA@gHB, tanh
        g_pX (stream, P, HAo, AC, ADP);                    // P @ HA'
        g_upd1(stream, AC, T, gDW, DAo, ADP, NARC);
        g_upd2(stream, DA, gDB, DAo, ADP, NARC);

        g_pXT(stream, P, DT, T, TDP);                      // dr = P^T @ DT
        g_pX (stream, P, HT, AC, TDP);                     // hr = P @ HT
        g_upd1(stream, AC, T, gHRW, HTo, TDP, NTAG);
        g_upd2(stream, HT, gHRB, HTo, TDP, NTAG);
        g_pX (stream, P, HTo, AC, TDP);
        g_upd1(stream, AC, T, gDRW, DTo, TDP, NTAG);
        g_upd2(stream, DT, gDRB, DTo, TDP, NTAG);

        swp(HA, HAo); swp(DA, DAo); swp(HT, HTo); swp(DT, DTo);
    }

    // final bilinear with layer-2 weights -> d_out
    float* out = (float*)d_out;
    const float* W2 = bilW + 2L * 501 * 501;
    g_bil1(stream, HA, W2, T);
    g_bil2(stream, T, DA, out, bilb + 2);

    const long ntag = (long)BB * SS * NTAG;
    copy_tag_kernel<<<(int)((ntag + 255) / 256), 256, 0, stream>>>(HT, out + SZP, ntag);
    copy_tag_kernel<<<(int)((ntag + 255) / 256), 256, 0, stream>>>(DT, out + SZP + ntag, ntag);
    finalize_kernel<<<1, 32, 0, stream>>>(SC, out + SZP + 2 * ntag);
}